// MultiEmbedding_18786186952924
// MI455X (gfx1250) — compile-verified
//
#include <hip/hip_runtime.h>
#include <hip/hip_bf16.h>

// MultiEmbedding: out[n,:] = sum_{l<8} weight[l, x[n,l], :]
// Memory-bound: 256MB output stream dominates; 32MB weight table is L2-resident
// (192MB L2), each row reused ~64x. CDNA5 path: TDM gather-mode
// tensor_load_to_lds pulls the 8 indexed rows per token into LDS
// (double-buffered, TENSORcnt-paced); the wave sums from LDS and streams the
// result with non-temporal 128-bit stores so the output write stream does not
// evict the weight working set from L2.

typedef __attribute__((ext_vector_type(4))) float        v4f;
typedef __attribute__((ext_vector_type(4))) unsigned int u32x4;
typedef __attribute__((ext_vector_type(8))) unsigned int u32x8;

#define WAVES   4               // waves per workgroup (wave32)
#define GRIDX   4096            // fixed launch grid -> compile-time wave stride
#define STRIDE  (GRIDX * WAVES) // 16384 global waves
#define LVLS    8               // L
#define KSZ     1024            // codebook size per level
#define DDIM    1024            // token dim
#define CHUNK   256             // columns gathered per TDM op (1KB/row)
#define CPT     (DDIM / CHUNK)  // chunks per token = 4 (power of two)
#define TILEF   (LVLS * CHUNK)  // floats per LDS tile = 2048 (8KB)

__global__ __launch_bounds__(WAVES * 32)
void multi_embed_tdm_kernel(const int* __restrict__ x,
                            const float* __restrict__ w,
                            float* __restrict__ out, int N)
{
    // 4 waves * 2 buffers * 8KB = 64KB LDS per workgroup (5 WGs / WGP by LDS)
    __shared__ float smem[WAVES * 2 * TILEF];

    const int lane  = threadIdx.x & 31;
    const int wave  = threadIdx.x >> 5;
    const int gwave = blockIdx.x * WAVES + wave;

    // STRIDE is a power-of-two compile-time constant -> shifts, no int division
    const int nTok  = (gwave < N) ? ((N - gwave + STRIDE - 1) / STRIDE) : 0;
    const int items = nTok * CPT;
    if (items == 0) return;

    // Workgroup-relative LDS byte addresses of this wave's two tile buffers
    // (low 32 bits of the generic address == LDS byte offset).
    unsigned int ldsBase[2];
    ldsBase[0] = (unsigned int)(size_t)(&smem[(wave * 2 + 0) * TILEF]);
    ldsBase[1] = (unsigned int)(size_t)(&smem[(wave * 2 + 1) * TILEF]);

    // Invariant D# group1: data_size=4B; tensor 8192x1024 f32, row stride 1024
    // elements; tile = 8 gathered rows x 256 cols; workgroup_mask=0 (no cluster).
    u32x8 g1;
    g1.s0 = 2u << 16;                                 // data_size=2 -> 4 bytes
    g1.s1 = ((unsigned)DDIM & 0xFFFFu) << 16;         // tensor_dim0[15:0]
    g1.s2 = ((unsigned)(LVLS * KSZ) & 0xFFFFu) << 16; // dim0 hi=0 | tensor_dim1 lo
    g1.s3 = ((unsigned)CHUNK) << 16;                  // dim1 hi=0 | tile_dim0
    g1.s4 = (unsigned)LVLS;                           // tile_dim1 = #valid indices
    g1.s5 = (unsigned)DDIM;                           // tensor_dim0_stride
    g1.s6 = 0u;
    g1.s7 = 0u;

    unsigned int rows[LVLS];   // row index into the flattened 8192x1024 table

    for (int it = 0; it <= items; ++it) {
        // ---- produce: issue TDM gather for item `it` ----
        if (it < items) {
            const int tok   = it >> 2;       // it / CPT
            const int chunk = it & 3;        // it % CPT   (wave-uniform)
            const int n     = gwave + tok * STRIDE;
            if (chunk == 0) {
#pragma unroll
                for (int l = 0; l < LVLS; ++l) {
                    int idx = __builtin_amdgcn_readfirstlane(x[n * LVLS + l]);
                    rows[l] = (unsigned int)(l * KSZ + idx);  // strictly increasing
                }
            }
            const unsigned long long gaddr =
                (unsigned long long)(size_t)w +
                (unsigned long long)(chunk * (CHUNK * 4));    // column offset

            u32x4 g0, g2, g3;
            // D# group0: count=1, gather_mode=1, 32-bit indices, lds/global addr, type=2
            g0.x = 0xC0000001u;
            g0.y = ldsBase[it & 1];
            g0.z = (unsigned int)gaddr;
            g0.w = (unsigned int)((gaddr >> 32) & 0x1FFFFFFull) | (2u << 30);
            // D# groups 2/3: the eight 32-bit row indices
            g2.x = rows[0]; g2.y = rows[1]; g2.z = rows[2]; g2.w = rows[3];
            g3.x = rows[4]; g3.y = rows[5]; g3.z = rows[6]; g3.w = rows[7];

            asm volatile("tensor_load_to_lds %0, %1, %2, %3"
                         :: "s"(g0), "s"(g1), "s"(g2), "s"(g3)
                         : "memory");
        }

        // ---- consume: item `it-1` once its TDM transfer has landed ----
        if (it > 0) {
            if (it < items)
                asm volatile("s_wait_tensorcnt 0x1" ::: "memory"); // in-order => it-1 done
            else
                asm volatile("s_wait_tensorcnt 0x0" ::: "memory"); // drain

            const int pit   = it - 1;
            const int tok   = pit >> 2;
            const int chunk = pit & 3;
            const int n     = gwave + tok * STRIDE;
            const float* buf = &smem[(wave * 2 + (pit & 1)) * TILEF];

            v4f a0 = {0.f, 0.f, 0.f, 0.f};
            v4f a1 = {0.f, 0.f, 0.f, 0.f};
#pragma unroll
            for (int l = 0; l < LVLS; ++l) {
                const v4f* r = (const v4f*)(buf + l * CHUNK);  // row l = 64 x v4f
                a0 += r[lane];        // cols [lane*4, lane*4+4)  -> balanced LDS banks
                a1 += r[32 + lane];   // cols [128+lane*4, ...)
            }
            v4f* o = (v4f*)(out + (size_t)n * DDIM + (size_t)(chunk * CHUNK));
            __builtin_nontemporal_store(a0, o + lane);        // 512B contiguous / wave
            __builtin_nontemporal_store(a1, o + 32 + lane);
        }
    }
}

extern "C" void kernel_launch(void* const* d_in, const int* in_sizes, int n_in,
                              void* d_out, int out_size, void* d_ws, size_t ws_size,
                              hipStream_t stream)
{
    const int*   x   = (const int*)d_in[0];     // [N, 8] indices
    const float* w   = (const float*)d_in[1];   // [8, 1024, 1024] f32
    float*       out = (float*)d_out;           // [N, 1024] f32

    const int N = in_sizes[0] / LVLS;           // 65536

    // GRIDX blocks x 4 waves = 16384 waves -> 4 tokens per wave at N=65536.
    multi_embed_tdm_kernel<<<GRIDX, WAVES * 32, 0, stream>>>(x, w, out, N);
}